// GCN_57397942944034
// MI455X (gfx1250) — compile-verified
//
#include <hip/hip_runtime.h>

typedef __attribute__((ext_vector_type(2))) float v2f;
typedef __attribute__((ext_vector_type(8))) float v8f;

#define WAVES_PER_BLOCK 4

// ---------------- degree / normalization ----------------

__global__ void gcn_deg_init(float* __restrict__ deg, int N) {
  int i = blockIdx.x * blockDim.x + threadIdx.x;
  if (i < N) deg[i] = 1.0f;  // self-loop contributes 1 to every node's degree
}

__global__ void gcn_deg_edge(const int* __restrict__ col, float* __restrict__ deg, int E) {
  int e = blockIdx.x * blockDim.x + threadIdx.x;
  if (e < E) unsafeAtomicAdd(&deg[col[e]], 1.0f);
}

__global__ void gcn_deg_to_dis(float* __restrict__ dis, int N) {
  int i = blockIdx.x * blockDim.x + threadIdx.x;
  if (i < N) {
    float d = dis[i];
    dis[i] = (d > 0.0f) ? rsqrtf(d) : 0.0f;
  }
}

// ---------------- graph aggregation ----------------

// agg[i] = dis[i]^2 * h[i]   (self-loop term; also initializes agg)
__global__ void gcn_self_loop(const float* __restrict__ dis, const float* __restrict__ h,
                              float* __restrict__ agg, int N, int C) {
  long long i = (long long)blockIdx.x * blockDim.x + threadIdx.x;
  long long total = (long long)N * C;
  if (i < total) {
    int node = (int)(i / C);
    float d = dis[node];
    agg[i] = d * d * h[i];
  }
}

// one wave32 per edge: agg[col] += dis[row]*dis[col] * h[row]
__global__ void gcn_edge_scatter(const int* __restrict__ ei, const float* __restrict__ dis,
                                 const float* __restrict__ h, float* __restrict__ agg,
                                 int E, int C) {
  int gw   = (int)(((long long)blockIdx.x * blockDim.x + threadIdx.x) >> 5);
  int lane = threadIdx.x & 31;
  if (gw >= E) return;
  int r = ei[gw];          // source
  int c = ei[E + gw];      // target
  float w = dis[r] * dis[c];
  const float* hp = h + (long long)r * C;
  float*       ap = agg + (long long)c * C;
  for (int ch = lane; ch < C; ch += 32)
    unsafeAtomicAdd(ap + ch, w * hp[ch]);
}

// ---------------- WMMA fp32 GEMM ----------------
// out[r, col] = inA[r,:]·WA[col,:] (+ inB[r,:]·WB[col,:]) + bA[col] (+ bB[col]) (, relu)
// A-tile (16 x IC) staged in LDS per block; each wave owns one 16x16 output tile.
// Uses V_WMMA_F32_16X16X4_F32 (exact fp32 — workload is scatter/bandwidth-bound,
// so the precision-preserving fp32 matrix path costs nothing vs fp16).
//
// Lane layouts per CDNA5 ISA 7.12.2 (32-bit, 16x4 A / 4x16 B, K stepped by 4):
//   A: lane L holds A[L&15][k + 2*(L>>4) + {0,1}]
//   B: lane L holds B[k + 2*(L>>4) + {0,1}][L&15]  (B = W^T, so W[col][k+koff+{0,1}])
//   C/D: lane L holds D[vg + 8*(L>>4)][L&15] in VGPR vg.
// GUARD=false: tile fully in-bounds (N%16==0 && OC%16==0) -> branch-free epilogue.
template <int IC, bool DUAL, bool RELU, bool GUARD>
__global__ __launch_bounds__(32 * WAVES_PER_BLOCK) void gcn_wmma_gemm(
    const float* __restrict__ inA, const float* __restrict__ WA, const float* __restrict__ bA,
    const float* __restrict__ inB, const float* __restrict__ WB, const float* __restrict__ bB,
    float* __restrict__ out, int nrows, int OC) {
  __shared__ float lds[(DUAL ? 2 : 1) * 16 * IC];
  const int tid  = threadIdx.x;
  const int lane = tid & 31;
  const int wave = tid >> 5;
  const int row0 = blockIdx.x << 4;
  const int col0 = (blockIdx.y * WAVES_PER_BLOCK + wave) << 4;

  float* Atile = lds;
  float* Btile = DUAL ? (lds + 16 * IC) : lds;

  // cooperative coalesced float4 staging of the 16-row activation tile(s)
  constexpr int vecPerRow = IC >> 2;
  constexpr int totalVec  = vecPerRow << 4;
  for (int idx = tid; idx < totalVec; idx += 32 * WAVES_PER_BLOCK) {
    const int r = idx / vecPerRow;
    const int v = idx - r * vecPerRow;
    const int node = row0 + r;
    if (GUARD && node >= nrows) {
      float4 z = {0.f, 0.f, 0.f, 0.f};
      ((float4*)Atile)[idx] = z;
      if constexpr (DUAL) ((float4*)Btile)[idx] = z;
    } else {
      ((float4*)Atile)[idx] = ((const float4*)(inA + (size_t)node * IC))[v];
      if constexpr (DUAL)
        ((float4*)Btile)[idx] = ((const float4*)(inB + (size_t)node * IC))[v];
    }
  }
  __syncthreads();

  if (col0 >= OC) return;  // wave-uniform: drop fully-empty column tiles

  const int  nidx = lane & 15;             // M for A-frag, N for B/C/D-frag
  const int  koff = (lane >> 4) << 1;      // 0 or 2
  const int  col  = col0 + nidx;
  // GUARD: clamp so OOB lanes load valid (but unused) weight rows; their garbage
  // only lands in D columns that are never stored.
  const bool colOK = GUARD ? (col < OC) : true;
  const int  colC  = GUARD ? (colOK ? col : (OC - 1)) : col;
  const float* wAp = WA + (size_t)colC * IC + koff;
  const float* wBp = DUAL ? (WB + (size_t)colC * IC + koff) : WA;
  const float* ar  = Atile + nidx * IC + koff;
  const float* br  = Btile + nidx * IC + koff;

  constexpr int KSTEPS = IC / 4;
  constexpr int BATCH  = DUAL ? (KSTEPS < 4 ? KSTEPS : 4)
                              : (KSTEPS < 8 ? KSTEPS : 8);

  v8f acc = {};
#pragma unroll 1
  for (int kb = 0; kb < KSTEPS; kb += BATCH) {
    v2f fa[BATCH], fw[BATCH];
    v2f ga[DUAL ? BATCH : 1], gw[DUAL ? BATCH : 1];
    // issue all loads for this batch up-front (one wait covers the group;
    // next batch's loads overlap the WMMA pipeline)
#pragma unroll
    for (int i = 0; i < BATCH; ++i) {
      const int k = (kb + i) * 4;
      fa[i] = *(const v2f*)(ar + k);
      fw[i] = *(const v2f*)(wAp + k);
      if constexpr (DUAL) {
        ga[i] = *(const v2f*)(br + k);
        gw[i] = *(const v2f*)(wBp + k);
      }
    }
#pragma unroll
    for (int i = 0; i < BATCH; ++i) {
      acc = __builtin_amdgcn_wmma_f32_16x16x4_f32(false, fa[i], false, fw[i],
                                                  (short)0, acc, false, false);
      if constexpr (DUAL)
        acc = __builtin_amdgcn_wmma_f32_16x16x4_f32(false, ga[i], false, gw[i],
                                                    (short)0, acc, false, false);
    }
  }

  const int mb = (lane >> 4) << 3;
  if constexpr (GUARD) {
    float bias = 0.f;
    if (colOK) {
      bias = bA[col];
      if constexpr (DUAL) bias += bB[col];
    }
#pragma unroll
    for (int vg = 0; vg < 8; ++vg) {
      const int r = row0 + mb + vg;
      if (colOK && r < nrows) {
        float val = acc[vg] + bias;
        if constexpr (RELU) val = val > 0.f ? val : 0.f;
        out[(size_t)r * OC + col] = val;
      }
    }
  } else {
    float bias = bA[col];
    if constexpr (DUAL) bias += bB[col];
    float* op = out + (size_t)(row0 + mb) * OC + col;
#pragma unroll
    for (int vg = 0; vg < 8; ++vg) {
      float val = acc[vg] + bias;
      if constexpr (RELU) val = val > 0.f ? val : 0.f;
      op[(size_t)vg * OC] = val;
    }
  }
}

// ---------------- host orchestration ----------------

template <int IC, bool DUAL, bool RELU>
static inline void launch_gemm(const float* inA, const float* WA, const float* bA,
                               const float* inB, const float* WB, const float* bB,
                               float* out, int N, int OC, hipStream_t stream) {
  const int rowTiles = (N + 15) / 16;
  const int gy = (((OC + 15) / 16) + WAVES_PER_BLOCK - 1) / WAVES_PER_BLOCK;
  const dim3 grid(rowTiles, gy), blk(32 * WAVES_PER_BLOCK);
  if ((N & 15) == 0 && (OC & 15) == 0)
    gcn_wmma_gemm<IC, DUAL, RELU, false><<<grid, blk, 0, stream>>>(
        inA, WA, bA, inB, WB, bB, out, N, OC);
  else
    gcn_wmma_gemm<IC, DUAL, RELU, true><<<grid, blk, 0, stream>>>(
        inA, WA, bA, inB, WB, bB, out, N, OC);
}

extern "C" void kernel_launch(void* const* d_in, const int* in_sizes, int n_in,
                              void* d_out, int out_size, void* d_ws, size_t ws_size,
                              hipStream_t stream) {
  (void)n_in; (void)out_size; (void)ws_size;
  const float* x    = (const float*)d_in[0];
  const int*   ei   = (const int*)d_in[1];   // [2,E]: [0:E)=row(src), [E:2E)=col(dst)
  const float* w1   = (const float*)d_in[2];
  const float* b1   = (const float*)d_in[3];
  const float* w2   = (const float*)d_in[4];
  const float* b2   = (const float*)d_in[5];
  const float* c1wl = (const float*)d_in[6];
  const float* c1bl = (const float*)d_in[7];
  const float* c1wr = (const float*)d_in[8];
  const float* c1br = (const float*)d_in[9];
  const float* c2wl = (const float*)d_in[10];
  const float* c2bl = (const float*)d_in[11];
  const float* c2wr = (const float*)d_in[12];
  const float* c2br = (const float*)d_in[13];
  const float* w3   = (const float*)d_in[14];
  const float* b3   = (const float*)d_in[15];
  const float* w4   = (const float*)d_in[16];
  const float* b4   = (const float*)d_in[17];

  const int IN_CH = 256;
  const int N = in_sizes[0] / IN_CH;
  const int E = in_sizes[1] / 2;
  const size_t n = (size_t)N;

  // workspace layout (floats), peak = 321*N floats ~= 128.4 MB
  float* ws   = (float*)d_ws;
  float* dis  = ws;              // N
  float* h2   = ws + 1 * n;      // 32N
  float* agg1 = ws + 33 * n;     // 32N
  float* h3   = ws + 65 * n;     // 64N
  float* agg2 = ws + 129 * n;    // 64N
  float* h4   = ws + 193 * n;    // 128N (ends at 321N)
  float* h1   = ws + 193 * n;    // 16N  -- dead before h4 is written
  float* h5   = ws + 1 * n;      // 192N -- reuses h2..agg2 region after they die

  const int eltBlocks  = (N + 255) / 256;
  const int edgeBlocks = (E + 255) / 256;
  const int scatBlocks = (int)(((long long)E * 32 + 255) / 256);

  // h1 = x @ w1^T + b1            [N,256] -> [N,16]
  launch_gemm<256, false, false>(x, w1, b1, nullptr, nullptr, nullptr, h1, N, 16, stream);
  // h2 = h1 @ w2^T + b2           [N,16] -> [N,32]
  launch_gemm<16, false, false>(h1, w2, b2, nullptr, nullptr, nullptr, h2, N, 32, stream);

  // degree + normalization
  gcn_deg_init<<<eltBlocks, 256, 0, stream>>>(dis, N);
  gcn_deg_edge<<<edgeBlocks, 256, 0, stream>>>(ei + E, dis, E);
  gcn_deg_to_dis<<<eltBlocks, 256, 0, stream>>>(dis, N);

  // conv1: agg1 = segsum(norm * h2[row]) ; h3 = relu(h2@c1wl^T + agg1@c1wr^T + biases)
  gcn_self_loop<<<(int)((n * 32 + 255) / 256), 256, 0, stream>>>(dis, h2, agg1, N, 32);
  gcn_edge_scatter<<<scatBlocks, 256, 0, stream>>>(ei, dis, h2, agg1, E, 32);
  launch_gemm<32, true, true>(h2, c1wl, c1bl, agg1, c1wr, c1br, h3, N, 64, stream);

  // conv2: C=64 -> [N,128]
  gcn_self_loop<<<(int)((n * 64 + 255) / 256), 256, 0, stream>>>(dis, h3, agg2, N, 64);
  gcn_edge_scatter<<<scatBlocks, 256, 0, stream>>>(ei, dis, h3, agg2, E, 64);
  launch_gemm<64, true, true>(h3, c2wl, c2bl, agg2, c2wr, c2br, h4, N, 128, stream);

  // h5 = h4 @ w3^T + b3           [N,128] -> [N,192]
  launch_gemm<128, false, false>(h4, w3, b3, nullptr, nullptr, nullptr, h5, N, 192, stream);
  // out = h5 @ w4^T + b4          [N,192] -> [N,20]
  launch_gemm<192, false, false>(h5, w4, b4, nullptr, nullptr, nullptr, (float*)d_out, N, 20, stream);
}